// FindSimilarIntentSess_84670985274059
// MI455X (gfx1250) — compile-verified
//
#include <hip/hip_runtime.h>
#include <hip/hip_bf16.h>
#include <math.h>

typedef __attribute__((ext_vector_type(2))) float v2f;
typedef __attribute__((ext_vector_type(8))) float v8f;

#define D        128
#define KSTEPS   32                 // D / 4 (one WMMA consumes K=4)
#define NW       8                  // waves per block (256 threads, wave32)
#define ITER     64                 // 16-col B tiles per block
#define SPAN     (ITER * 16)        // 1024 pool columns per block
#define BPAD     132                // LDS B-stage row pitch (conflict-free for 4k+2*hi reads)
#define NEGF     (-3.0e38f)
#define EPS_TERM (128.0f * 1e-6f)   // sum(x*x + eps) = sumsq + D*eps

// ---------------- streaming softmax + top-3 state ----------------
struct RowStat { float m, s, t0, t1, t2; int i0, i1, i2; };

__device__ __forceinline__ void stat_init(RowStat& st) {
    st.m = NEGF; st.s = 0.0f;
    st.t0 = st.t1 = st.t2 = NEGF;
    st.i0 = st.i1 = st.i2 = 0;
}

__device__ __forceinline__ void top_push(RowStat& st, float x, int c) {
    if (x > st.t0)      { st.t2 = st.t1; st.i2 = st.i1; st.t1 = st.t0; st.i1 = st.i0; st.t0 = x; st.i0 = c; }
    else if (x > st.t1) { st.t2 = st.t1; st.i2 = st.i1; st.t1 = x;  st.i1 = c; }
    else if (x > st.t2) { st.t2 = x;  st.i2 = c; }
}

__device__ __forceinline__ void stat_push(RowStat& st, float x, int c) {
    if (x > NEGF) {
        if (x > st.m) { st.s = st.s * __expf(st.m - x) + 1.0f; st.m = x; }
        else          { st.s += __expf(x - st.m); }
        top_push(st, x, c);
    }
}

__device__ __forceinline__ void stat_merge(RowStat& a, const RowStat& b) {
    if (b.m > NEGF) {
        if (b.m > a.m) { a.s = a.s * __expf(a.m - b.m) + b.s; a.m = b.m; }
        else           { a.s += b.s * __expf(b.m - a.m); }
        top_push(a, b.t0, b.i0);
        top_push(a, b.t1, b.i1);
        top_push(a, b.t2, b.i2);
    }
}

__device__ __forceinline__ void stat_store(float* p, const RowStat& st) {
    p[0] = st.m;  p[1] = st.s;
    p[2] = st.t0; p[3] = st.t1; p[4] = st.t2;
    p[5] = __int_as_float(st.i0);
    p[6] = __int_as_float(st.i1);
    p[7] = __int_as_float(st.i2);
}

__device__ __forceinline__ RowStat stat_load(const float* p) {
    RowStat st;
    st.m = p[0];  st.s = p[1];
    st.t0 = p[2]; st.t1 = p[3]; st.t2 = p[4];
    st.i0 = __float_as_int(p[5]);
    st.i1 = __float_as_int(p[6]);
    st.i2 = __float_as_int(p[7]);
    return st;
}

// ---------------- kernel 1: 1/sqrt(sumsq + D*eps), wave per row ----------------
__global__ void __launch_bounds__(256)
norms_kernel(const float* __restrict__ sess, const float* __restrict__ pool,
             float* __restrict__ rn_s, float* __restrict__ rn_p, int B, int P) {
    const int wv = threadIdx.x >> 5;
    const int ln = threadIdx.x & 31;
    const int row = blockIdx.x * NW + wv;
    if (row >= B + P) return;
    const float* src = (row < B) ? (sess + (size_t)row * D)
                                 : (pool + (size_t)(row - B) * D);
    const float4 x = ((const float4*)src)[ln];           // 32 lanes * 16B = full 512B row
    float ss = x.x * x.x + x.y * x.y + x.z * x.z + x.w * x.w;
    #pragma unroll
    for (int off = 16; off > 0; off >>= 1) ss += __shfl_xor(ss, off, 32);
    if (ln == 0) {
        const float r = rsqrtf(ss + EPS_TERM);
        if (row < B) rn_s[row] = r; else rn_p[row - B] = r;
    }
}

// ---------------- kernel 2: fused WMMA cos-sim + online softmax + top-3 ----------
// Block covers 128 sess rows (wave w owns rows [wv*16, wv*16+16)); all 8 waves share
// each 16-column B tile staged coalesced into double-buffered LDS.
__global__ void __launch_bounds__(256)
cossim_topk_kernel(const float* __restrict__ sess, const float* __restrict__ pool,
                   const float* __restrict__ rn_s, const float* __restrict__ rn_p,
                   float* __restrict__ partials, int B, int P, int nchunks) {
    __shared__ float bstage[2][16][BPAD];   // double-buffered B tile (16 pool rows x 128 K)
    __shared__ float tile[NW][16][17];      // per-wave 16x16 C tile (transpose for row scan)

    const int tid = threadIdx.x;
    const int wv  = tid >> 5;
    const int ln  = tid & 31;
    const int hi  = ln >> 4;                // half-wave selects K pair (fp32 A/B layout)
    const int nl  = ln & 15;
    const int rowblock  = blockIdx.x * (NW * 16);   // 128 sess rows per block
    const int chunkbase = blockIdx.y * SPAN;

    // A tile: 16 rows x 128 K in registers, pre-scaled by 1/||sess_row||.
    // Per-lane M = nl constant; VGPR pair holds K = 4k + 2*hi + {0,1}.
    const int   arow = rowblock + wv * 16 + nl;
    const float rs   = rn_s[arow];
    const float* aptr = sess + (size_t)arow * D + 2 * hi;
    v2f a[KSTEPS];
    #pragma unroll
    for (int k = 0; k < KSTEPS; ++k) {
        v2f t = *(const v2f*)(aptr + 4 * k);
        a[k] = t * rs;
    }

    RowStat st; stat_init(st);
    const int scanbase = hi * 8;            // this lane scans 8 of the 16 tile columns

    // Staging map: thread t copies 8 floats of pool row (t>>4) at col offset (t&15)*8.
    const int srow = tid >> 4;
    const int soff = (tid & 15) * 8;

    // Prologue: stage tile 0
    {
        const int col = chunkbase + srow;
        const int pr  = (col < P) ? col : (P - 1);
        const float4* src = (const float4*)(pool + (size_t)pr * D + soff);
        *(float4*)(&bstage[0][srow][soff])     = src[0];
        *(float4*)(&bstage[0][srow][soff + 4]) = src[1];
    }

    for (int it = 0; it < ITER; ++it) {
        const int buf     = it & 1;
        const int colbase = chunkbase + it * 16;

        __syncthreads();                    // bstage[buf] staged & visible block-wide

        // Stage tile it+1 into the other buffer (global latency overlaps WMMA below)
        if (it + 1 < ITER) {
            const int col = colbase + 16 + srow;
            const int pr  = (col < P) ? col : (P - 1);
            const float4* src = (const float4*)(pool + (size_t)pr * D + soff);
            {   // prefetch two tiles ahead
                const int pc  = col + 16;
                const int pfr = (pc < P) ? pc : (P - 1);
                __builtin_prefetch(pool + (size_t)pfr * D + soff, 0, 0);
            }
            const float4 x0 = src[0];
            const float4 x1 = src[1];
            *(float4*)(&bstage[buf ^ 1][srow][soff])     = x0;
            *(float4*)(&bstage[buf ^ 1][srow][soff + 4]) = x1;
        }

        // 16x16 raw dot tile: 32 chained f32 WMMAs, B fragments from LDS
        const float* bbase = &bstage[buf][nl][2 * hi];
        v8f acc0 = {0.f,0.f,0.f,0.f,0.f,0.f,0.f,0.f};
        v8f acc1 = {0.f,0.f,0.f,0.f,0.f,0.f,0.f,0.f};
        #pragma unroll
        for (int k = 0; k < KSTEPS; k += 2) {
            v2f b0 = *(const v2f*)(bbase + 4 * k);
            v2f b1 = *(const v2f*)(bbase + 4 * (k + 1));
            acc0 = __builtin_amdgcn_wmma_f32_16x16x4_f32(false, a[k],     false, b0,
                                                         (short)0, acc0, false, false);
            acc1 = __builtin_amdgcn_wmma_f32_16x16x4_f32(false, a[k + 1], false, b1,
                                                         (short)0, acc1, false, false);
        }
        const v8f acc = acc0 + acc1;

        // scale by 1/||pool_col|| (C layout: per-lane column N = nl constant)
        const int   bcol = colbase + nl;
        const float cp   = (bcol < P) ? rn_p[bcol] : 0.0f;

        __syncthreads();
        #pragma unroll
        for (int v = 0; v < 8; ++v) {
            tile[wv][v + 8 * hi][nl] = (bcol < P) ? acc[v] * cp : NEGF;  // M=v+8*hi, N=nl
        }
        __syncthreads();

        // row-major scan: lane owns row nl of its wave, scans its 8-column half
        #pragma unroll
        for (int c = 0; c < 8; ++c) {
            const int cc = scanbase + c;
            stat_push(st, tile[wv][nl][cc], colbase + cc);
        }
    }

    // Merge the two half-wave scan partials via cross-lane shuffles (no LDS needed:
    // each wave exclusively owns its 16 rows).
    RowStat o;
    o.m  = __shfl_down(st.m,  16, 32);
    o.s  = __shfl_down(st.s,  16, 32);
    o.t0 = __shfl_down(st.t0, 16, 32);
    o.t1 = __shfl_down(st.t1, 16, 32);
    o.t2 = __shfl_down(st.t2, 16, 32);
    o.i0 = __shfl_down(st.i0, 16, 32);
    o.i1 = __shfl_down(st.i1, 16, 32);
    o.i2 = __shfl_down(st.i2, 16, 32);
    if (ln < 16) {
        stat_merge(st, o);
        stat_store(partials + ((size_t)(rowblock + wv * 16 + nl) * nchunks + blockIdx.y) * 8, st);
    }
}

// ---------------- kernel 3: merge chunk partials per row ----------------
__global__ void __launch_bounds__(256)
merge_kernel(const float* __restrict__ partials, float* __restrict__ finals,
             int B, int nchunks) {
    const int row = blockIdx.x * blockDim.x + threadIdx.x;
    if (row >= B) return;
    RowStat st; stat_init(st);
    const float* p = partials + (size_t)row * nchunks * 8;
    for (int c = 0; c < nchunks; ++c) {
        RowStat o = stat_load(p + (size_t)c * 8);
        stat_merge(st, o);
    }
    stat_store(finals + (size_t)row * 8, st);
}

// ---------------- kernel 4: finalize outputs ----------------
__global__ void __launch_bounds__(128)
finalize_kernel(const float* __restrict__ pool, const float* __restrict__ finals,
                float* __restrict__ out, int B, int P) {
    const int row = blockIdx.x;
    const int d   = threadIdx.x;
    const float* f = finals + (size_t)row * 8;
    const float m = f[0], s = f[1];
    const float t0 = f[2], t1 = f[3], t2 = f[4];
    const int i0 = __float_as_int(f[5]);
    const int i1 = __float_as_int(f[6]);
    const int i2 = __float_as_int(f[7]);

    // cos_topk (pre re-softmax) = exp(t - rowmax) / Z ; t0 >= t1 >= t2
    const float inv = 1.0f / s;
    const float p0 = __expf(t0 - m) * inv;
    const float p1 = __expf(t1 - m) * inv;
    const float p2 = __expf(t2 - m) * inv;
    // second softmax over k=3 (max is p0)
    const float e0 = 1.0f;
    const float e1 = __expf(p1 - p0);
    const float e2 = __expf(p2 - p0);
    const float wsum = e0 + e1 + e2;
    const float w0 = e0 / wsum, w1 = e1 / wsum, w2 = e2 / wsum;

    const float g0 = pool[(size_t)i0 * D + d];
    const float g1 = pool[(size_t)i1 * D + d];
    const float g2 = pool[(size_t)i2 * D + d];

    const size_t o1 = (size_t)B * D;            // cos_topk offset
    const size_t o2 = o1 + (size_t)B * 3;       // sess_topk offset
    out[(size_t)row * D + d] = w0 * g0 + w1 * g1 + w2 * g2;       // neighbor_sess
    out[o2 + ((size_t)row * 3 + 0) * D + d] = g0;                 // sess_topk
    out[o2 + ((size_t)row * 3 + 1) * D + d] = g1;
    out[o2 + ((size_t)row * 3 + 2) * D + d] = g2;
    if (d == 0) {
        out[o1 + (size_t)row * 3 + 0] = w0;                       // cos_topk
        out[o1 + (size_t)row * 3 + 1] = w1;
        out[o1 + (size_t)row * 3 + 2] = w2;
    }
}

// ---------------- launch ----------------
extern "C" void kernel_launch(void* const* d_in, const int* in_sizes, int n_in,
                              void* d_out, int out_size, void* d_ws, size_t ws_size,
                              hipStream_t stream) {
    const float* sess = (const float*)d_in[0];
    const float* pool = (const float*)d_in[1];
    float* out = (float*)d_out;

    const int B = in_sizes[0] / D;              // 2048
    const int P = in_sizes[1] / D;              // 100000
    const int nchunks = (P + SPAN - 1) / SPAN;  // 98

    float* ws       = (float*)d_ws;
    float* rn_s     = ws;                                   // [B]
    float* rn_p     = rn_s + B;                             // [P]
    float* partials = rn_p + P;                             // [B * nchunks * 8]
    float* finals   = partials + (size_t)B * nchunks * 8;   // [B * 8]

    const int totalRows = B + P;
    norms_kernel<<<dim3((totalRows + NW - 1) / NW), 256, 0, stream>>>(
        sess, pool, rn_s, rn_p, B, P);

    cossim_topk_kernel<<<dim3(B / (NW * 16), nchunks), 256, 0, stream>>>(
        sess, pool, rn_s, rn_p, partials, B, P, nchunks);

    merge_kernel<<<dim3((B + 255) / 256), 256, 0, stream>>>(partials, finals, B, nchunks);

    finalize_kernel<<<dim3(B), 128, 0, stream>>>(pool, finals, out, B, P);
}